// Model_40750649704595
// MI455X (gfx1250) — compile-verified
//
#include <hip/hip_runtime.h>
#include <math.h>

typedef __attribute__((ext_vector_type(2))) float v2f;
typedef __attribute__((ext_vector_type(8))) float v8f;
typedef int v4i_vs __attribute__((vector_size(16)));   // matches async-LDS builtin pointee

#define T_LEN    4096
#define K_WIN    12
#define NGROUPS  66
#define NTHREADS 256
#define NWAVES   8

#if defined(__has_builtin)
#  if __has_builtin(__builtin_amdgcn_global_load_async_to_lds_b128) && \
      __has_builtin(__builtin_amdgcn_s_wait_asynccnt)
#    define USE_ASYNC_LDS 1
#  endif
#  if __has_builtin(__builtin_amdgcn_permlanex16)
#    define HAVE_PERMX16 1
#  endif
#endif
#ifndef USE_ASYNC_LDS
#  define USE_ASYNC_LDS 0
#endif
#ifndef HAVE_PERMX16
#  define HAVE_PERMX16 0
#endif

#define AS1 __attribute__((address_space(1)))
#define AS3 __attribute__((address_space(3)))

// Window sizes (static, derived from T=4096 exactly as reference _window_sizes)
__constant__ int c_WS[K_WIN]   = {10, 17, 31, 56, 100, 177, 316, 562, 1000, 1778, 3162, 4096};
__constant__ int c_NSEG[K_WIN] = {409, 240, 132, 73, 40, 23, 12, 7, 4, 2, 1, 1};

// Flattened worklist: 66 groups of up to 16 segments, ordered by descending
// chunk count so the long serial scans land on different waves.
__constant__ signed char c_GK[NGROUPS] = {
    11, 10, 9, 8, 7, 6,
    5, 5,
    4, 4, 4,
    3, 3, 3, 3, 3,
    2, 2, 2, 2, 2, 2, 2, 2, 2,
    1, 1, 1, 1, 1, 1, 1, 1, 1, 1, 1, 1, 1, 1, 1,
    0, 0, 0, 0, 0, 0, 0, 0, 0, 0, 0, 0, 0,
    0, 0, 0, 0, 0, 0, 0, 0, 0, 0, 0, 0, 0
};
__constant__ short c_GS[NGROUPS] = {
    0, 0, 0, 0, 0, 0,
    0, 16,
    0, 16, 32,
    0, 16, 32, 48, 64,
    0, 16, 32, 48, 64, 80, 96, 112, 128,
    0, 16, 32, 48, 64, 80, 96, 112, 128, 144, 160, 176, 192, 208, 224,
    0, 16, 32, 48, 64, 80, 96, 112, 128, 144, 160, 176, 192,
    208, 224, 240, 256, 272, 288, 304, 320, 336, 352, 368, 384, 400
};

// Cross-half-wave exchange (lane i <-> lane i^16) without touching LDS:
// v_permlanex16_b32 with identity selectors -> pure VALU on the carry path.
__device__ __forceinline__ float xhalf(float v) {
#if HAVE_PERMX16
    const int s = __float_as_int(v);
    const int t = __builtin_amdgcn_permlanex16(s, s, 0x76543210, 0xFEDCBA98, false, false);
    return __int_as_float(t);
#else
    return __shfl_xor(v, 16, 32);
#endif
}

// One 16-row x 16-segment prefix-scan step:
//   Z(16x16) = L(16x16, lower-tri ones) x DEV(16x16), chained as 4 WMMAs over K.
// MASKED=true only for the final (partial) chunk of a segment.
template <bool MASKED>
__device__ __forceinline__ void rs_chunk(const float* __restrict__ row,
                                         const v2f A[4],
                                         int base_idx,   // base0 + cb + 2*half
                                         int row_idx,    // cb + 2*half (dev-row of b.x at kk=0)
                                         int wm2,        // w - 2
                                         float mi,
                                         float& carry, float& zmax, float& zmin,
                                         float& sum2)
{
    v8f acc = {0.f, 0.f, 0.f, 0.f, 0.f, 0.f, 0.f, 0.f};
    float psum = 0.0f;   // this lane's 8 dev values of its column (half the chunk)
#pragma unroll
    for (int kk = 0; kk < 4; ++kk) {
        // B layout (32-bit 4x16) mirrors A's half split: row r = 4*kk + v + 2*half
        const int idx = base_idx + 4 * kk;
        const float x0 = row[idx];
        const float x1 = row[idx + 1];
        const float x2 = row[idx + 2];
        float d0 = (x1 - x0) - mi;
        float d1 = (x2 - x1) - mi;
        if (MASKED) {
            const int r = row_idx + 4 * kk;
            d0 = (r     <= wm2) ? d0 : 0.0f;
            d1 = (r + 1 <= wm2) ? d1 : 0.0f;
        }
        v2f b;
        b.x = d0;
        b.y = d1;
        sum2 += d0 * d0 + d1 * d1;
        psum += d0 + d1;
        acc = __builtin_amdgcn_wmma_f32_16x16x4_f32(
            false, A[kk], false, b, (short)0, acc, false, false);
    }
    // C/D layout: lane col holds rows m = 8*half + j of its column.
    // Padded rows replicate the last valid prefix -> harmless to max/min.
#pragma unroll
    for (int j = 0; j < 8; ++j) {
        const float z = acc[j] + carry;
        zmax = fmaxf(zmax, z);
        zmin = fminf(zmin, z);
    }
    // Chunk total = sum of all 16 dev rows of the column; assembled from the
    // two half-wave partial sums (VALU permlane, no LDS on the carry chain).
    carry += psum + xhalf(psum);
}

__global__ __launch_bounds__(NTHREADS)
void hurst_rs_kernel(const float* __restrict__ x, float* __restrict__ out) {
    __shared__ float row[T_LEN + 32];   // +pad so chunk tail reads stay in-bounds
    __shared__ float rsacc[K_WIN];

    const int tid = threadIdx.x;
    const float* xrow = x + (size_t)blockIdx.x * T_LEN;

    // ---- Stage one row (16KB) into LDS: HBM read exactly once ----
#if USE_ASYNC_LDS
    {
        AS1 v4i_vs* g = (AS1 v4i_vs*)(unsigned long long)(uintptr_t)xrow;
        AS3 v4i_vs* l = (AS3 v4i_vs*)(unsigned int)(uintptr_t)(&row[0]);
        for (int i = tid; i < T_LEN / 4; i += NTHREADS) {
            __builtin_amdgcn_global_load_async_to_lds_b128(g + i, l + i, 0, 0);
        }
    }
#else
    {
        const float4* src = (const float4*)xrow;
        float4* dst = (float4*)row;
        for (int i = tid; i < T_LEN / 4; i += NTHREADS) dst[i] = src[i];
    }
#endif
    if (tid < 32) row[T_LEN + tid] = 0.0f;
    if (tid < K_WIN) rsacc[tid] = 0.0f;
#if USE_ASYNC_LDS
    __builtin_amdgcn_s_wait_asynccnt(0);
#endif
    __syncthreads();

    const int lane = tid & 31;
    const int wave = __builtin_amdgcn_readfirstlane(tid >> 5);  // force SGPR control flow
    const int half = lane >> 4;   // half-wave index (K/M split in WMMA layouts)
    const int col  = lane & 15;   // segment column / matrix row index m

    // A operands: columns of the 16x16 lower-triangular ones matrix L.
    // A layout (32-bit 16x4): lane m (+16 for upper K pair), VGPR v -> K = v + 2*half.
    v2f A[4];
#pragma unroll
    for (int kk = 0; kk < 4; ++kk) {
        A[kk].x = (col >= 4 * kk + 2 * half + 0) ? 1.0f : 0.0f;
        A[kk].y = (col >= 4 * kk + 2 * half + 1) ? 1.0f : 0.0f;
    }

    // ---- Work items: each = (window k, group of 16 segments); all scalars uniform ----
    for (int gi = wave; gi < NGROUPS; gi += NWAVES) {
        const int k    = __builtin_amdgcn_readfirstlane((int)c_GK[gi]);
        const int w    = __builtin_amdgcn_readfirstlane(c_WS[k]);
        const int nseg = __builtin_amdgcn_readfirstlane(c_NSEG[k]);
        const int s0   = __builtin_amdgcn_readfirstlane((int)c_GS[gi]);

        int seg = s0 + col;
        const bool valid = (lane < 16) && (seg < nseg);
        if (seg > nseg - 1) seg = nseg - 1;      // clamp fake columns (recompute, discard)
        const int base0 = seg * w;

        // mean increment = mean of d (telescoping), so S^2 = sum(dev^2)/(w-2)
        const float mi = (row[base0 + w - 1] - row[base0]) / (float)(w - 1);

        float carry = 0.0f;
        float zmax = -3.4e38f, zmin = 3.4e38f, sum2 = 0.0f;

        const int nfull = (w - 1) >> 4;          // chunks with all 16 dev rows valid
        const int rem   = (w - 1) & 15;

        int base_idx = base0 + 2 * half;
        for (int c = 0; c < nfull; ++c) {
            rs_chunk<false>(row, A, base_idx, 0, w - 2, mi,
                            carry, zmax, zmin, sum2);
            base_idx += 16;
        }
        if (rem) {
            rs_chunk<true>(row, A, base_idx, (nfull << 4) + 2 * half, w - 2, mi,
                           carry, zmax, zmin, sum2);
        }

        // combine the two half-wave row groups of each column
        zmax = fmaxf(zmax, xhalf(zmax));
        zmin = fminf(zmin, xhalf(zmin));
        sum2 +=            xhalf(sum2);

        const float R = zmax - zmin;
        const float S = sqrtf(sum2 / (float)(w - 2));   // ddof=1 over (w-1) samples
        float rsv = valid ? (R / (S + 1e-6f)) : 0.0f;

        // wave-wide sum of this group's R/S values, one LDS atomic per group
#pragma unroll
        for (int off = 1; off < 32; off <<= 1) rsv += __shfl_xor(rsv, off, 32);
        if (lane == 0) atomicAdd(&rsacc[k], rsv);
    }
    __syncthreads();

    // ---- 12-point log-log regression -> H ----
    if (tid == 0) {
        float X[K_WIN], Y[K_WIN];
        float mx = 0.0f, my = 0.0f;
#pragma unroll
        for (int k = 0; k < K_WIN; ++k) {
            X[k] = log10f((float)c_WS[k] + 1e-6f);
            Y[k] = log10f(rsacc[k] / (float)c_NSEG[k] + 1e-6f);
            mx += X[k];
            my += Y[k];
        }
        mx *= (1.0f / K_WIN);
        my *= (1.0f / K_WIN);
        float num = 0.0f, den = 0.0f;
#pragma unroll
        for (int k = 0; k < K_WIN; ++k) {
            const float xc = X[k] - mx;
            num += (Y[k] - my) * xc;
            den += xc * xc;
        }
        float H = num / (den + 1e-6f);
        if (!__builtin_isfinite(H)) H = 1.0f;
        out[blockIdx.x] = H;   // SHIFT == 0.0
    }
}

extern "C" void kernel_launch(void* const* d_in, const int* in_sizes, int n_in,
                              void* d_out, int out_size, void* d_ws, size_t ws_size,
                              hipStream_t stream) {
    const float* x = (const float*)d_in[0];
    float* out = (float*)d_out;
    const int B = in_sizes[0] / T_LEN;   // 8192 rows, one block per row
    hipLaunchKernelGGL(hurst_rs_kernel, dim3(B), dim3(NTHREADS), 0, stream, x, out);
}